// granular_BERT_48704929137381
// MI455X (gfx1250) — compile-verified
//
#include <hip/hip_runtime.h>
#include <math.h>
#include <stdint.h>

// ---------------- problem constants ----------------
#define BN  512   // batch
#define LCC 90    // context length
#define LTT 10    // target length
#define DD  768   // bert dim
#define HH  128   // lstm hidden
#define H2  256   // 2H
#define H4  512   // 4H

typedef float f32x2 __attribute__((ext_vector_type(2)));
typedef float f32x8 __attribute__((ext_vector_type(8)));

// CDNA5 full-precision WMMA: D(16x16,f32) = A(16x4,f32) * B(4x16,f32) + C
__device__ __forceinline__ f32x8 wmma4(f32x2 a, f32x2 b, f32x8 c) {
  return __builtin_amdgcn_wmma_f32_16x16x4_f32(false, a, false, b, (short)0, c,
                                               false, false);
}

__device__ __forceinline__ float sigf(float x) { return 1.0f / (1.0f + expf(-x)); }

// LDS byte offset of a generic pointer to __shared__ (flat aperture keeps the
// LDS offset in addr[31:0] on CDNA5).
__device__ __forceinline__ unsigned lds_off(const void* p) {
  return (unsigned)(uintptr_t)p;
}

// Per-lane async copy of 16 bytes global -> LDS (tracked by ASYNCcnt).
__device__ __forceinline__ void async_g2lds_b128(unsigned ldsByteOff,
                                                 const void* gaddr) {
  asm volatile("global_load_async_to_lds_b128 %0, %1, off"
               :: "v"(ldsByteOff), "v"(gaddr)
               : "memory");
}
__device__ __forceinline__ void async_wait0() {
  asm volatile("s_wait_asynccnt 0" ::: "memory");
}

// ====================================================================
// Generic GEMM: C[M,N] = A[M,K] * op(B) (+ bias[n]).
// TRANSB==0 : B stored [K,N] row-major.   TRANSB==1 : B stored [N,K] row-major.
// Requirements: M%16==0, N%128==0, K%64==0.
// grid = (M/16, N/128), block = 256 (8 waves), one 16x16 C tile per wave.
// The 16xK A tile is shared by all 8 waves: staged 64 columns at a time into
// LDS with GLOBAL_LOAD_ASYNC_TO_LDS_B128 (256 lanes x 16B = 16x64 floats).
// TRANSB is a template constant so both inner loops are branch-free and B
// addressing strength-reduces to pointer increments.
// ====================================================================
template <int TRANSB>
__global__ void gemm_wmma_f32(const float* __restrict__ A,
                              const float* __restrict__ Bm,
                              const float* __restrict__ bias,
                              float* __restrict__ C,
                              int M, int N, int K) {
  __shared__ float aS[16 * 64];
  const int tid  = threadIdx.x;
  const int lane = tid & 31;
  const int wave = tid >> 5;
  const int tm   = blockIdx.x;
  const int tn   = blockIdx.y * 8 + wave;
  const int half = lane >> 4;        // which K/M half of the wave
  const int mr   = lane & 15;        // A row / B col / D col within tile
  const long colB = (long)tn * 16 + mr;

  // async staging role of this thread: one 16B chunk of the 16x64 A tile
  const int srow = tid >> 4;         // 0..15  (A row within tile)
  const int sseg = tid & 15;         // 0..15  (16B segment within row chunk)
  const float* gsrcA = A + ((long)tm * 16 + srow) * K + sseg * 4;
  const unsigned ldsA = lds_off(&aS[srow * 64 + sseg * 4]);

  // per-lane bases (loop bodies below are pure pointer arithmetic)
  const float* pA = &aS[mr * 64 + half * 2];               // LDS A-frag base
  const float* pB = TRANSB ? (Bm + colB * K + half * 2)    // row of B^T
                           : (Bm + colB + (long)(half * 2) * N);

  f32x8 acc = {0.f,0.f,0.f,0.f,0.f,0.f,0.f,0.f};
  for (int kc = 0; kc < K; kc += 64) {
    async_g2lds_b128(ldsA, gsrcA + kc);
    if (TRANSB) {
      __builtin_prefetch(pB + kc + 64, 0, 1);              // global_prefetch_b8
    }
    async_wait0();
    __syncthreads();
    if (TRANSB) {
#pragma unroll
      for (int k0 = 0; k0 < 64; k0 += 4) {
        f32x2 a = {pA[k0], pA[k0 + 1]};
        f32x2 b = {pB[kc + k0], pB[kc + k0 + 1]};
        acc = wmma4(a, b, acc);
      }
    } else {
      const float* pBk = pB + (long)kc * N;
#pragma unroll
      for (int k0 = 0; k0 < 64; k0 += 4) {
        f32x2 a = {pA[k0], pA[k0 + 1]};
        f32x2 b = {pBk[0], pBk[N]};
        acc = wmma4(a, b, acc);
        pBk += 4 * (long)N;
      }
    }
    __syncthreads();           // safe to overwrite aS next chunk
  }
  const float bv = bias ? bias[tn * 16 + mr] : 0.0f;
#pragma unroll
  for (int v = 0; v < 8; ++v) {
    C[((long)tm * 16 + half * 8 + v) * N + (long)tn * 16 + mr] = acc[v] + bv;
  }
}

// ====================================================================
// G1CT[b] = CW[b] (90x768) @ target[b]^T (768x10)  -> [90,10]
// grid = B, block = 192 (6 waves of 16 rows, N padded 10->16).
// Out-of-range rows/cols are merely CLAMPED: the polluted products land only
// in output cells that are never stored, so the inner loop is branch-free.
// ====================================================================
__global__ void g1ct_wmma(const float* __restrict__ CW,
                          const float* __restrict__ tgt,
                          float* __restrict__ G1) {
  const int b    = blockIdx.x;
  const int lane = threadIdx.x & 31;
  const int wave = threadIdx.x >> 5;
  const int half = lane >> 4;
  const int mr   = lane & 15;
  const int row  = wave * 16 + mr;                 // LC row (may exceed 89)
  const int ra   = (row < LCC) ? row : 0;          // clamp to stay in-bounds
  const int rb   = (mr  < LTT) ? mr  : 0;
  const float* Ab = CW  + (long)b * LCC * DD + (long)ra * DD + half * 2;
  const float* Tb = tgt + (long)b * LTT * DD + (long)rb * DD + half * 2;

  f32x8 acc = {0.f,0.f,0.f,0.f,0.f,0.f,0.f,0.f};
  for (int k0 = 0; k0 < DD; k0 += 4) {
    f32x2 a  = {Ab[k0], Ab[k0 + 1]};
    f32x2 bb = {Tb[k0], Tb[k0 + 1]};
    acc = wmma4(a, bb, acc);
  }
#pragma unroll
  for (int v = 0; v < 8; ++v) {
    const int r = wave * 16 + half * 8 + v;
    if (r < LCC && mr < LTT)
      G1[((long)b * LCC + r) * LTT + mr] = acc[v];
  }
}

// ====================================================================
// LSTM recurrence (one direction per blockIdx.y).
// Workgroup = 16 batch rows, loops all T steps; gates via f32 WMMA.
// Xg = x @ Wih^T + b, precomputed, layout [B, T, 512] (time in original order).
// out[b, t, dir*128 + n] with bwd writing position T-1-t.
// grid = (B/16, 2), block = 256 (8 waves).
// ====================================================================
__global__ void lstm_rec(const float* __restrict__ Xg_f,
                         const float* __restrict__ Xg_b,
                         const float* __restrict__ Whh2,   // [2,512,128]
                         float* __restrict__ out,          // [B,T,256]
                         int T) {
  __shared__ float hS[16 * 128];
  __shared__ float cS[16 * 128];
  __shared__ float gS[16 * 512];
  const int dir = blockIdx.y;
  const int b0  = blockIdx.x * 16;
  const float* Xg  = dir ? Xg_b : Xg_f;
  const float* Whh = Whh2 + (long)dir * H4 * HH;
  const int tid  = threadIdx.x;
  const int lane = tid & 31;
  const int wave = tid >> 5;
  const int half = lane >> 4;
  const int mr   = lane & 15;
  const float* pA = &hS[mr * HH + half * 2];       // LDS h-frag base

  for (int i = tid; i < 16 * 128; i += 256) { hS[i] = 0.f; cS[i] = 0.f; }
  __syncthreads();

  for (int t = 0; t < T; ++t) {
    const int ti = dir ? (T - 1 - t) : t;
    // ---- gates = Xg[:, ti, :] + h @ Whh^T  (wave owns 64 gate columns) ----
#pragma unroll
    for (int nt = 0; nt < 4; ++nt) {
      const int n0 = wave * 64 + nt * 16;
      const float* pB = Whh + (long)(n0 + mr) * HH + half * 2;  // Whh[n][k]
      f32x8 acc;
#pragma unroll
      for (int v = 0; v < 8; ++v)
        acc[v] = Xg[((long)(b0 + half * 8 + v) * T + ti) * H4 + n0 + mr];
      for (int k0 = 0; k0 < HH; k0 += 4) {
        f32x2 a  = {pA[k0], pA[k0 + 1]};
        f32x2 bb = {pB[k0], pB[k0 + 1]};
        acc = wmma4(a, bb, acc);
      }
#pragma unroll
      for (int v = 0; v < 8; ++v)
        gS[(half * 8 + v) * H4 + n0 + mr] = acc[v];
    }
    __syncthreads();
    // ---- cell update (pytorch gate order i,f,g,o) ----
    for (int i = tid; i < 16 * 128; i += 256) {
      const int m = i >> 7, n = i & 127;
      const float ig = gS[m * H4 + n];
      const float fg = gS[m * H4 + 128 + n];
      const float gg = gS[m * H4 + 256 + n];
      const float og = gS[m * H4 + 384 + n];
      const float c  = sigf(fg) * cS[i] + sigf(ig) * tanhf(gg);
      const float h  = sigf(og) * tanhf(c);
      cS[i] = c; hS[i] = h;
      out[((long)(b0 + m) * T + ti) * H2 + dir * HH + n] = h;
    }
    __syncthreads();
  }
}

// ====================================================================
// out[b,i,n] = sum_j G(b,i,j) * V[b,j,n]   (tiny-K batched product)
// G element = G[b*900 + i*grs + j*gks];  V is [B, Kd, 256].
// grid = (B, R), block = 256 (n)
// ====================================================================
__global__ void bmm_gv(const float* __restrict__ G, const float* __restrict__ V,
                       float* __restrict__ out, int R, int Kd, int grs, int gks) {
  const int b = blockIdx.x, i = blockIdx.y, n = threadIdx.x;
  const float* g = G + (long)b * LCC * LTT + (long)i * grs;
  const float* v = V + (long)b * Kd * H2 + n;
  float s = 0.f;
  for (int j = 0; j < Kd; ++j) s += g[(long)j * gks] * v[(long)j * H2];
  out[((long)b * R + i) * H2 + n] = s;
}

// G2[b,i,j] = 1 / (1 + ||s2[b,i]-t2[b,j]||)
__global__ void euclid_k(const float* __restrict__ Sm, const float* __restrict__ Tm,
                         float* __restrict__ G2) {
  const long idx = (long)blockIdx.x * blockDim.x + threadIdx.x;
  if (idx >= (long)BN * LCC * LTT) return;
  const int j = (int)(idx % LTT);
  const long r = idx / LTT;
  const int i = (int)(r % LCC);
  const int b = (int)(r / LCC);
  const float* c = Sm + ((long)b * LCC + i) * H2;
  const float* t = Tm + ((long)b * LTT + j) * H2;
  float d2 = 0.f;
  for (int k = 0; k < H2; ++k) { const float d = c[k] - t[k]; d2 += d * d; }
  G2[idx] = 1.0f / (1.0f + sqrtf(d2));
}

// P[b,i,:] = softmax_j( g[b,i] . g[b,j] );  grid=(B,L), block=128
__global__ void attn_softmax(const float* __restrict__ g, float* __restrict__ P, int L) {
  const int b = blockIdx.x, i = blockIdx.y, j = threadIdx.x;
  __shared__ float sc[128];
  __shared__ float red[2];
  float d = 0.f;
  if (j < L) {
    const float* gi = g + ((long)b * L + i) * H2;
    const float* gj = g + ((long)b * L + j) * H2;
    for (int k = 0; k < H2; ++k) d += gi[k] * gj[k];
  }
  sc[j] = (j < L) ? d : -1e30f;
  __syncthreads();
  if (j == 0) { float m = -1e30f; for (int x = 0; x < L; ++x) m = fmaxf(m, sc[x]); red[0] = m; }
  __syncthreads();
  const float e = (j < L) ? expf(sc[j] - red[0]) : 0.f;
  sc[j] = e;
  __syncthreads();
  if (j == 0) { float s = 0.f; for (int x = 0; x < L; ++x) s += sc[x]; red[1] = s; }
  __syncthreads();
  if (j < L) P[((long)b * L + i) * L + j] = e / red[1];
}

// w[b,k] = sum_j (sum_i P[b,i,j]) * g[b,j,k];  grid=B, block=256
__global__ void attn_pool(const float* __restrict__ P, const float* __restrict__ g,
                          float* __restrict__ w, int L) {
  const int b = blockIdx.x, tid = threadIdx.x;
  __shared__ float cs[96];
  if (tid < L) {
    float s = 0.f;
    for (int i = 0; i < L; ++i) s += P[((long)b * L + i) * L + tid];
    cs[tid] = s;
  }
  __syncthreads();
  float acc = 0.f;
  for (int j = 0; j < L; ++j) acc += cs[j] * g[((long)b * L + j) * H2 + tid];
  w[(long)b * H2 + tid] = acc;
}

// x[b,i] = Y[b,i,:] . w[b,:]
__global__ void rowdot(const float* __restrict__ Y, const float* __restrict__ w,
                       float* __restrict__ x, int L) {
  const int idx = blockIdx.x * 256 + threadIdx.x;
  if (idx >= BN * L) return;
  const int b = idx / L;
  const float* y  = Y + (long)idx * H2;
  const float* ww = w + (long)b * H2;
  float s = 0.f;
  for (int k = 0; k < H2; ++k) s += y[k] * ww[k];
  x[idx] = s;
}

// cols[b,j] = sum_i exp(tanh(x[b,i] + bias[j])); partial[b] = sum_j cols[b,j]
__global__ void alpha_colsum(const float* __restrict__ x, const float* __restrict__ bias,
                             float* __restrict__ cols, float* __restrict__ partial, int L) {
  const int b = blockIdx.x, j = threadIdx.x;     // block = 128
  __shared__ float xs[96];
  __shared__ float red[128];
  if (j < L) xs[j] = x[(long)b * L + j];
  __syncthreads();
  float s = 0.f;
  if (j < L) {
    const float bj = bias[j];
    for (int i = 0; i < L; ++i) s += expf(tanhf(xs[i] + bj));
    cols[(long)b * L + j] = s;
  }
  red[j] = s;
  __syncthreads();
  for (int off = 64; off > 0; off >>= 1) { if (j < off) red[j] += red[j + off]; __syncthreads(); }
  if (j == 0) partial[b] = red[0];
}

// deterministic sum of 512 partials -> *S
__global__ void reduce512(const float* __restrict__ part, float* __restrict__ S) {
  __shared__ float sm[512];
  const int t = threadIdx.x;
  sm[t] = part[t];
  __syncthreads();
  for (int off = 256; off > 0; off >>= 1) { if (t < off) sm[t] += sm[t + off]; __syncthreads(); }
  if (t == 0) *S = sm[0];
}

// out[b,k] = (1/S) * sum_j cols[b,j] * g[b,j,k];  grid=B, block=256
__global__ void alpha_pool(const float* __restrict__ cols, const float* __restrict__ g,
                           const float* __restrict__ S, float* __restrict__ out, int L) {
  const int b = blockIdx.x, k = threadIdx.x;
  __shared__ float cs[96];
  if (k < L) cs[k] = cols[(long)b * L + k];
  __syncthreads();
  const float inv = 1.0f / S[0];
  float s = 0.f;
  for (int j = 0; j < L; ++j) s += cs[j] * g[((long)b * L + j) * H2 + k];
  out[(long)b * H2 + k] = s * inv;
}

// final FC (3 x 4H) + softmax over classes
__global__ void fc_softmax(const float* __restrict__ Gc, const float* __restrict__ Gt,
                           const float* __restrict__ fw, const float* __restrict__ fb,
                           float* __restrict__ out) {
  const int b = blockIdx.x * 64 + threadIdx.x;
  if (b >= BN) return;
  float z[3];
#pragma unroll
  for (int c = 0; c < 3; ++c) {
    float s = fb[c];
    const float* w = fw + c * 2 * H2;
    for (int k = 0; k < H2; ++k) s += w[k] * Gc[(long)b * H2 + k];
    for (int k = 0; k < H2; ++k) s += w[H2 + k] * Gt[(long)b * H2 + k];
    z[c] = s;
  }
  const float m = fmaxf(z[0], fmaxf(z[1], z[2]));
  const float e0 = expf(z[0] - m), e1 = expf(z[1] - m), e2 = expf(z[2] - m);
  const float inv = 1.f / (e0 + e1 + e2);
  out[b * 3 + 0] = e0 * inv;
  out[b * 3 + 1] = e1 * inv;
  out[b * 3 + 2] = e2 * inv;
}

// ====================================================================
extern "C" void kernel_launch(void* const* d_in, const int* in_sizes, int n_in,
                              void* d_out, int out_size, void* d_ws, size_t ws_size,
                              hipStream_t stream) {
  (void)in_sizes; (void)n_in; (void)out_size; (void)ws_size;
  const float* ctx     = (const float*)d_in[0];
  const float* tgt     = (const float*)d_in[1];
  const float* W       = (const float*)d_in[2];
  const float* W_a     = (const float*)d_in[3];
  const float* b_a     = (const float*)d_in[4];
  const float* W_b     = (const float*)d_in[5];
  const float* b_b     = (const float*)d_in[6];
  const float* fc_w    = (const float*)d_in[7];
  const float* fc_b    = (const float*)d_in[8];
  const float* l1s_Wih = (const float*)d_in[9];
  const float* l1s_Whh = (const float*)d_in[10];
  const float* l1s_b   = (const float*)d_in[11];
  const float* l1t_Wih = (const float*)d_in[12];
  const float* l1t_Whh = (const float*)d_in[13];
  const float* l1t_b   = (const float*)d_in[14];
  const float* l2s_Wih = (const float*)d_in[15];
  const float* l2s_Whh = (const float*)d_in[16];
  const float* l2s_b   = (const float*)d_in[17];
  const float* l2t_Wih = (const float*)d_in[18];
  const float* l2t_Whh = (const float*)d_in[19];
  const float* l2t_b   = (const float*)d_in[20];
  float* ws  = (float*)d_ws;
  float* out = (float*)d_out;

  // ---- workspace map (floats); aliasing noted ----
  float* XGf   = ws + 0L;           // dir0 gate pre-acts  (also holds CW early)
  float* XGb   = ws + 23592960L;    // dir1 gate pre-acts
  float* CW    = ws + 0L;           // aliases XG arena (dead before Xg GEMMs)
  float* s_out = ws + 47185920L;    // [B,90,256]  (later s_out3 aliases this)
  float* t_out = ws + 58982400L;    // [B,10,256]  (later t_out3 aliases this)
  float* s_out2= ws + 60293120L;
  float* t_out2= ws + 72089600L;
  float* sin_  = ws + 73400320L;    // s_in2 / s_in3 / Y_c
  float* tin_  = ws + 85196800L;    // t_in2 / t_in3 / Y_t
  float* G1    = ws + 86507520L;
  float* G2    = ws + 86968320L;
  float* AC    = ws + 87429120L;    // [B,90,90]
  float* AT    = ws + 91576320L;    // [B,10,10]
  float* wt    = ws + 91627520L;
  float* wc    = ws + 91758592L;
  float* xc    = ws + 91889664L;
  float* xt    = ws + 91935744L;
  float* colsC = ws + 91940864L;
  float* colsT = ws + 91986944L;
  float* Ssum  = ws + 91992064L;    // [2]
  float* Gc    = ws + 91992066L;
  float* Gt    = ws + 92123138L;
  float* partC = ws + 92254210L;    // [512]
  float* partT = ws + 92254722L;
  float* s_out3 = s_out;
  float* t_out3 = t_out;

  auto gemm = [&](const float* A, const float* Bm, const float* bias, float* C,
                  int M, int N, int K, int transB) {
    dim3 g(M / 16, N / 128);
    if (transB)
      gemm_wmma_f32<1><<<g, 256, 0, stream>>>(A, Bm, bias, C, M, N, K);
    else
      gemm_wmma_f32<0><<<g, 256, 0, stream>>>(A, Bm, bias, C, M, N, K);
  };
  const dim3 recGrid(BN / 16, 2);

  // 1) CW = ctx @ W ; G1CT = CW @ tgt^T
  gemm(ctx, W, nullptr, CW, BN * LCC, DD, DD, 0);
  g1ct_wmma<<<BN, 192, 0, stream>>>(CW, tgt, G1);

  // 2) s_out = biLSTM_l1s(ctx)
  gemm(ctx, l1s_Wih,            l1s_b,      XGf, BN * LCC, H4, DD, 1);
  gemm(ctx, l1s_Wih + H4 * DD,  l1s_b + H4, XGb, BN * LCC, H4, DD, 1);
  lstm_rec<<<recGrid, 256, 0, stream>>>(XGf, XGb, l1s_Whh, s_out, LCC);

  // 3) t_out = biLSTM_l1t(tgt)
  gemm(tgt, l1t_Wih,            l1t_b,      XGf, BN * LTT, H4, DD, 1);
  gemm(tgt, l1t_Wih + H4 * DD,  l1t_b + H4, XGb, BN * LTT, H4, DD, 1);
  lstm_rec<<<recGrid, 256, 0, stream>>>(XGf, XGb, l1t_Whh, t_out, LTT);

  // 4) s_in2 = G1 @ t_out ; t_in2 = G1^T @ s_out
  bmm_gv<<<dim3(BN, LCC), 256, 0, stream>>>(G1, t_out, sin_, LCC, LTT, LTT, 1);
  bmm_gv<<<dim3(BN, LTT), 256, 0, stream>>>(G1, s_out, tin_, LTT, LCC, 1, LTT);

  // 5) s_out2 = biLSTM_l2s(s_in2) ; t_out2 = biLSTM_l2t(t_in2)
  gemm(sin_, l2s_Wih,           l2s_b,      XGf, BN * LCC, H4, H2, 1);
  gemm(sin_, l2s_Wih + H4 * H2, l2s_b + H4, XGb, BN * LCC, H4, H2, 1);
  lstm_rec<<<recGrid, 256, 0, stream>>>(XGf, XGb, l2s_Whh, s_out2, LCC);
  gemm(tin_, l2t_Wih,           l2t_b,      XGf, BN * LTT, H4, H2, 1);
  gemm(tin_, l2t_Wih + H4 * H2, l2t_b + H4, XGb, BN * LTT, H4, H2, 1);
  lstm_rec<<<recGrid, 256, 0, stream>>>(XGf, XGb, l2t_Whh, t_out2, LTT);

  // 6) G2 = euclid(s_out2, t_out2) ; s_in3 = G2 @ t_out2 ; t_in3 = G2^T @ s_out2
  euclid_k<<<(BN * LCC * LTT + 255) / 256, 256, 0, stream>>>(s_out2, t_out2, G2);
  bmm_gv<<<dim3(BN, LCC), 256, 0, stream>>>(G2, t_out2, sin_, LCC, LTT, LTT, 1);
  bmm_gv<<<dim3(BN, LTT), 256, 0, stream>>>(G2, s_out2, tin_, LTT, LCC, 1, LTT);

  // 7) s_out3 = biLSTM_l2s(s_in3) ; t_out3 = biLSTM_l2t(t_in3)
  gemm(sin_, l2s_Wih,           l2s_b,      XGf, BN * LCC, H4, H2, 1);
  gemm(sin_, l2s_Wih + H4 * H2, l2s_b + H4, XGb, BN * LCC, H4, H2, 1);
  lstm_rec<<<recGrid, 256, 0, stream>>>(XGf, XGb, l2s_Whh, s_out3, LCC);
  gemm(tin_, l2t_Wih,           l2t_b,      XGf, BN * LTT, H4, H2, 1);
  gemm(tin_, l2t_Wih + H4 * H2, l2t_b + H4, XGb, BN * LTT, H4, H2, 1);
  lstm_rec<<<recGrid, 256, 0, stream>>>(XGf, XGb, l2t_Whh, t_out3, LTT);

  // 8) self-attention pools: wt from t_out3, wc from s_out3
  attn_softmax<<<dim3(BN, LTT), 128, 0, stream>>>(t_out3, AT, LTT);
  attn_pool<<<BN, 256, 0, stream>>>(AT, t_out3, wt, LTT);
  attn_softmax<<<dim3(BN, LCC), 128, 0, stream>>>(s_out3, AC, LCC);
  attn_pool<<<BN, 256, 0, stream>>>(AC, s_out3, wc, LCC);

  // 9) x_c = (s_out3 @ W_a) . wt ;  x_t = (t_out3 @ W_b) . wc
  gemm(s_out3, W_a, nullptr, sin_, BN * LCC, H2, H2, 0);
  rowdot<<<(BN * LCC + 255) / 256, 256, 0, stream>>>(sin_, wt, xc, LCC);
  gemm(t_out3, W_b, nullptr, tin_, BN * LTT, H2, H2, 0);
  rowdot<<<(BN * LTT + 255) / 256, 256, 0, stream>>>(tin_, wc, xt, LTT);

  // 10) global-sum exp(tanh(.)) attention (deterministic two-stage reduce)
  alpha_colsum<<<BN, 128, 0, stream>>>(xc, b_a, colsC, partC, LCC);
  alpha_colsum<<<BN, 128, 0, stream>>>(xt, b_b, colsT, partT, LTT);
  reduce512<<<1, 512, 0, stream>>>(partC, Ssum + 0);
  reduce512<<<1, 512, 0, stream>>>(partT, Ssum + 1);
  alpha_pool<<<BN, 256, 0, stream>>>(colsC, s_out3, Ssum + 0, Gc, LCC);
  alpha_pool<<<BN, 256, 0, stream>>>(colsT, t_out3, Ssum + 1, Gt, LTT);

  // 11) final FC + softmax
  fc_softmax<<<(BN + 63) / 64, 64, 0, stream>>>(Gc, Gt, fc_w, fc_b, out);
}